// LoRARelationWiseWordSelectionHead_27788438405211
// MI455X (gfx1250) — compile-verified
//
#include <hip/hip_runtime.h>

#define B_ 8
#define S_ 1024
#define H_ 1024
#define KC 32          // K chunk
#define KP 36          // padded LDS row stride (floats): conflict-free, keeps 8B/16B alignment
#define CP 20          // padded coefficient row stride (floats): conflict-free, 16B aligned

typedef __attribute__((ext_vector_type(2))) float v2f;
typedef __attribute__((ext_vector_type(8))) float v8f;

// ---------------------------------------------------------------------------
// Pass 1: per (b,s):  c[j=r*4+l] = <x_row, U[:,r,l]>,  d[j] = <x_row, V[r,:,l]>
// ---------------------------------------------------------------------------
__global__ __launch_bounds__(256) void coeff_kernel(const float* __restrict__ x,
                                                    const float* __restrict__ U,
                                                    const float* __restrict__ V,
                                                    float* __restrict__ c,
                                                    float* __restrict__ d) {
    const int bs  = blockIdx.x;
    const int tid = threadIdx.x;
    const float* xrow = x + (size_t)bs * H_;

    float ac[8], ad[8];
#pragma unroll
    for (int j = 0; j < 8; ++j) { ac[j] = 0.0f; ad[j] = 0.0f; }

    float4 xv = *(const float4*)(xrow + tid * 4);
    const float* u0 = U + (size_t)tid * 32;
#pragma unroll
    for (int i = 0; i < 4; ++i) {
        const int h = tid * 4 + i;
        const float xs = (&xv.x)[i];
#pragma unroll
        for (int j = 0; j < 8; ++j) ac[j] += xs * u0[i * 8 + j];
        float4 v0 = *(const float4*)(V + (size_t)h * 4);            // r=0, l=0..3
        float4 v1 = *(const float4*)(V + (size_t)(H_ + h) * 4);     // r=1, l=0..3
#pragma unroll
        for (int l = 0; l < 4; ++l) {
            ad[l]     += xs * (&v0.x)[l];
            ad[4 + l] += xs * (&v1.x)[l];
        }
    }

#pragma unroll
    for (int j = 0; j < 8; ++j) {
#pragma unroll
        for (int mask = 16; mask >= 1; mask >>= 1) {
            ac[j] += __shfl_xor(ac[j], mask, 32);
            ad[j] += __shfl_xor(ad[j], mask, 32);
        }
    }

    __shared__ float wred[8][16];
    const int wave = tid >> 5, lane = tid & 31;
    if (lane == 0) {
#pragma unroll
        for (int j = 0; j < 8; ++j) { wred[wave][j] = ac[j]; wred[wave][8 + j] = ad[j]; }
    }
    __syncthreads();
    if (tid < 8) {
        float sc = 0.0f, sd = 0.0f;
#pragma unroll
        for (int w = 0; w < 8; ++w) { sc += wred[w][tid]; sd += wred[w][8 + tid]; }
        c[(size_t)bs * 8 + tid] = sc;
        d[(size_t)bs * 8 + tid] = sd;
    }
}

// ---------------------------------------------------------------------------
// Gv[l*4 + r*2 + r2] = <V[r,:,l], V[r2,:,l]>
// ---------------------------------------------------------------------------
__global__ __launch_bounds__(256) void vgram_kernel(const float* __restrict__ V,
                                                    float* __restrict__ gv) {
    const int tid = threadIdx.x;
    float a[16];
#pragma unroll
    for (int j = 0; j < 16; ++j) a[j] = 0.0f;
#pragma unroll
    for (int i = 0; i < 4; ++i) {
        const int h = tid * 4 + i;
        float4 v0 = *(const float4*)(V + (size_t)h * 4);
        float4 v1 = *(const float4*)(V + (size_t)(H_ + h) * 4);
#pragma unroll
        for (int l = 0; l < 4; ++l) {
            float a0 = (&v0.x)[l], a1 = (&v1.x)[l];
            a[l * 4 + 0] += a0 * a0;
            a[l * 4 + 1] += a0 * a1;
            a[l * 4 + 2] += a1 * a0;
            a[l * 4 + 3] += a1 * a1;
        }
    }
#pragma unroll
    for (int j = 0; j < 16; ++j)
#pragma unroll
        for (int mask = 16; mask >= 1; mask >>= 1) a[j] += __shfl_xor(a[j], mask, 32);

    __shared__ float wred[8][16];
    const int wave = tid >> 5, lane = tid & 31;
    if (lane == 0) {
#pragma unroll
        for (int j = 0; j < 16; ++j) wred[wave][j] = a[j];
    }
    __syncthreads();
    if (tid < 16) {
        float s = 0.0f;
#pragma unroll
        for (int w = 0; w < 8; ++w) s += wred[w][tid];
        gv[tid] = s;
    }
}

// ---------------------------------------------------------------------------
// Pass 2: per b, Gx = X X^T tile (128x128) once; per-l rank-4 correction on store.
// Double-buffered GLOBAL_LOAD_ASYNC_TO_LDS_B128 pipeline (ASYNCcnt), one
// barrier per K-chunk; 8 waves x (64x32 strip, 8 v8f accumulators).
// ---------------------------------------------------------------------------
__global__ __launch_bounds__(256) void gram_kernel(const float* __restrict__ x,
                                                   const float* __restrict__ c,
                                                   const float* __restrict__ d,
                                                   const float* __restrict__ gv,
                                                   float* __restrict__ out) {
    __shared__ float hS[2][128 * KP];
    __shared__ float hT[2][128 * KP];
    __shared__ float rcL[128 * CP];   // per S-row, per l: (c0, c1, d0, d1)
    __shared__ float ccL[128 * CP];   // per T-col, per l: (c0, c1, e0, e1)

    const int b  = blockIdx.z;
    const int s0 = blockIdx.x * 128;
    const int t0 = blockIdx.y * 128;

    const float* xb = x + (size_t)b * S_ * H_;
    const float* cb = c + (size_t)b * S_ * 8;
    const float* db = d + (size_t)b * S_ * 8;

    const int tid  = threadIdx.x;
    const int wave = tid >> 5;
    const int lane = tid & 31;
    const int m    = lane & 15;
    const int q    = lane >> 4;
    const int wm   = wave >> 2;   // 0..1  -> M offset wm*64
    const int wn   = wave & 3;    // 0..3  -> N offset wn*32

    // ---- stage row/col coefficients (ordering via first pipeline barrier) ----
    if (tid < 128) {
        const int r = tid;
        float4 cA = *(const float4*)(cb + (size_t)(s0 + r) * 8);
        float4 cB = *(const float4*)(cb + (size_t)(s0 + r) * 8 + 4);
        float4 dA = *(const float4*)(db + (size_t)(s0 + r) * 8);
        float4 dB = *(const float4*)(db + (size_t)(s0 + r) * 8 + 4);
#pragma unroll
        for (int l = 0; l < 4; ++l) {
            rcL[r * CP + l * 4 + 0] = (&cA.x)[l];
            rcL[r * CP + l * 4 + 1] = (&cB.x)[l];
            rcL[r * CP + l * 4 + 2] = (&dA.x)[l];
            rcL[r * CP + l * 4 + 3] = (&dB.x)[l];
        }
    } else {
        const int r = tid - 128;
        float4 cA = *(const float4*)(cb + (size_t)(t0 + r) * 8);
        float4 cB = *(const float4*)(cb + (size_t)(t0 + r) * 8 + 4);
        float4 dA = *(const float4*)(db + (size_t)(t0 + r) * 8);
        float4 dB = *(const float4*)(db + (size_t)(t0 + r) * 8 + 4);
#pragma unroll
        for (int l = 0; l < 4; ++l) {
            float c0 = (&cA.x)[l], c1 = (&cB.x)[l];
            float e0 = (&dA.x)[l] + gv[l * 4 + 0] * c0 + gv[l * 4 + 1] * c1;
            float e1 = (&dB.x)[l] + gv[l * 4 + 2] * c0 + gv[l * 4 + 3] * c1;
            ccL[r * CP + l * 4 + 0] = c0;
            ccL[r * CP + l * 4 + 1] = c1;
            ccL[r * CP + l * 4 + 2] = e0;
            ccL[r * CP + l * 4 + 3] = e1;
        }
    }

    // per-thread fill coordinates (1024 float4 per 128x32 tile; 4 per thread)
    int frow[4], fcol[4];
    unsigned gOffS[4], gOffT[4];
#pragma unroll
    for (int j = 0; j < 4; ++j) {
        int idx = tid + j * 256;
        frow[j] = idx >> 3;
        fcol[j] = (idx & 7) * 4;
        gOffS[j] = (unsigned)(((s0 + frow[j]) * H_ + fcol[j]) * 4);   // + kb*4 at issue
        gOffT[j] = (unsigned)(((t0 + frow[j]) * H_ + fcol[j]) * 4);
    }

    // async copy of one K-chunk into buffer `buf`
    auto issue_chunk = [&](int buf, int kb) {
#pragma unroll
        for (int j = 0; j < 4; ++j) {
            unsigned ldsS = (unsigned)(uintptr_t)&hS[buf][frow[j] * KP + fcol[j]];
            unsigned ldsT = (unsigned)(uintptr_t)&hT[buf][frow[j] * KP + fcol[j]];
            unsigned aS = gOffS[j] + (unsigned)(kb * 4);
            unsigned aT = gOffT[j] + (unsigned)(kb * 4);
            asm volatile("global_load_async_to_lds_b128 %0, %1, %2 offset:0"
                         :: "v"(ldsS), "v"(aS), "s"(xb) : "memory");
            asm volatile("global_load_async_to_lds_b128 %0, %1, %2 offset:0"
                         :: "v"(ldsT), "v"(aT), "s"(xb) : "memory");
        }
    };

    v8f acc[8];
#pragma unroll
    for (int t = 0; t < 8; ++t) acc[t] = (v8f)0.0f;

    issue_chunk(0, 0);   // prologue

    for (int kb = 0; kb < H_; kb += 2 * KC) {
#pragma unroll
        for (int p = 0; p < 2; ++p) {               // ping-pong, constant-folded
            const int cur = p, nxt = p ^ 1;
            const int kc  = kb + p * KC;

            asm volatile("s_wait_asynccnt 0x0" ::: "memory");  // own copies landed
            __syncthreads();                                    // everyone's landed; prev compute done

            if (kc + KC < H_) issue_chunk(nxt, kc + KC);        // overlap with compute

#pragma unroll
            for (int k = 0; k < KC / 4; ++k) {
                v2f a[4], bf[2];
#pragma unroll
                for (int i = 0; i < 4; ++i)
                    a[i] = *(const v2f*)&hS[cur][(wm * 64 + i * 16 + m) * KP + k * 4 + q * 2];
#pragma unroll
                for (int j = 0; j < 2; ++j)
                    bf[j] = *(const v2f*)&hT[cur][(wn * 32 + j * 16 + m) * KP + k * 4 + q * 2];
#pragma unroll
                for (int i = 0; i < 4; ++i)
#pragma unroll
                    for (int j = 0; j < 2; ++j)
                        acc[i * 2 + j] = __builtin_amdgcn_wmma_f32_16x16x4_f32(
                            false, a[i], false, bf[j], (short)0, acc[i * 2 + j], false, false);
            }
        }
    }

    // ---- epilogue: rank-4 correction per l, write 4 outputs ----
    float4 cc[2][4];
#pragma unroll
    for (int j = 0; j < 2; ++j)
#pragma unroll
        for (int l = 0; l < 4; ++l)
            cc[j][l] = *(const float4*)&ccL[(wn * 32 + j * 16 + m) * CP + l * 4];

#pragma unroll
    for (int i = 0; i < 4; ++i) {
#pragma unroll
        for (int v = 0; v < 8; ++v) {
            const int row = wm * 64 + i * 16 + 8 * q + v;   // C/D: M = v + 8q
#pragma unroll
            for (int l = 0; l < 4; ++l) {
                float4 rc = *(const float4*)&rcL[row * CP + l * 4];   // broadcast read
                const size_t ob = ((size_t)(b * 4 + l) * S_ + (s0 + row)) * S_ + t0;
#pragma unroll
                for (int j = 0; j < 2; ++j) {
                    float val = acc[i * 2 + j][v]
                              + rc.z * cc[j][l].x + rc.w * cc[j][l].y
                              + rc.x * cc[j][l].z + rc.y * cc[j][l].w;
                    out[ob + wn * 32 + j * 16 + m] = val;
                }
            }
        }
    }
}

extern "C" void kernel_launch(void* const* d_in, const int* in_sizes, int n_in,
                              void* d_out, int out_size, void* d_ws, size_t ws_size,
                              hipStream_t stream) {
    const float* x = (const float*)d_in[0];   // (B,S,H)
    const float* U = (const float*)d_in[1];   // (H,R,L)
    const float* V = (const float*)d_in[2];   // (R,H,L)
    float* out = (float*)d_out;               // (B,L,S,S)

    float* c  = (float*)d_ws;                 // B*S*8 floats
    float* d  = c + (size_t)B_ * S_ * 8;      // B*S*8 floats
    float* gv = d + (size_t)B_ * S_ * 8;      // 16 floats

    coeff_kernel<<<B_ * S_, 256, 0, stream>>>(x, U, V, c, d);
    vgram_kernel<<<1, 256, 0, stream>>>(V, gv);

    dim3 grid(S_ / 128, S_ / 128, B_);
    gram_kernel<<<grid, 256, 0, stream>>>(x, c, d, gv, out);
}